// SMPLModel_76115410420048
// MI455X (gfx1250) — compile-verified
//
#include <hip/hip_runtime.h>
#include <math.h>

#define PREG 200
#define NV   6890
#define KJ   24
#define KDIM 115          // 23 joints * 5 features
#define KPAD 116          // padded K (multiple of 4)
#define PPAD 208          // padded P (multiple of 16)
#define MDIM 20670        // 3*NV
#define MPAD 20672        // padded M (multiple of 16, and of 64 for 4-tile waves)
#define EPSF 1e-8f
#define CT   4            // column tiles per GEMM wave

typedef __attribute__((ext_vector_type(2))) float v2f;
typedef __attribute__((ext_vector_type(4))) float v4f;
typedef __attribute__((ext_vector_type(8))) float v8f;

// ---------------- prep: Qpad (PPAD x KPAD), row-major, zero padded ----------
__global__ void prep_q(const float* __restrict__ theta,
                       const float* __restrict__ beta2,
                       float* __restrict__ Qpad) {
    int idx = blockIdx.x * blockDim.x + threadIdx.x;   // over PPAD*KJ
    if (idx >= PPAD * KJ) return;
    int p = idx / KJ, j = idx % KJ;
    if (j == 0) { Qpad[p * KPAD + 115] = 0.f; return; }   // pad column
    float* dst = Qpad + p * KPAD + 5 * (j - 1);
    if (p >= PREG) { for (int f = 0; f < 5; ++f) dst[f] = 0.f; return; }
    const float* th = theta + ((size_t)p * KJ + j) * 3;
    float hx = th[0] * 0.5f, hy = th[1] * 0.5f, hz = th[2] * 0.5f;
    float n  = sqrtf(hx * hx + hy * hy + hz * hz);
    float w  = cosf(n);
    float s  = sinf(n) / (n + EPSF);
    dst[0] = w - 1.f;          // q_diff.w  (q_star = (1,0,0,0))
    dst[1] = hx * s;
    dst[2] = hy * s;
    dst[3] = hz * s;
    dst[4] = beta2[0];
}

// ---------------- prep: Rodrigues -> per (p,k) 3x4 [R | J] ------------------
__global__ void prep_rod(const float* __restrict__ theta,
                         const float* __restrict__ J,
                         float* __restrict__ Gmat) {
    int idx = blockIdx.x * blockDim.x + threadIdx.x;   // over PREG*KJ
    if (idx >= PREG * KJ) return;
    const float* th = theta + (size_t)idx * 3;
    float x = th[0], y = th[1], z = th[2];
    float tn = sqrtf(x * x + y * y + z * z);
    float inv = 1.f / (tn + EPSF);
    float ox = x * inv, oy = y * inv, oz = z * inv;
    float st = sinf(tn), ct = 1.f - cosf(tn);
    float* G = Gmat + (size_t)idx * 12;
    const float* Jp = J + (size_t)idx * 3;
    G[0]  = 1.f + ct * (-(oy * oy + oz * oz));
    G[1]  = st * (-oz) + ct * (ox * oy);
    G[2]  = st * ( oy) + ct * (ox * oz);
    G[3]  = Jp[0];
    G[4]  = st * ( oz) + ct * (ox * oy);
    G[5]  = 1.f + ct * (-(ox * ox + oz * oz));
    G[6]  = st * (-ox) + ct * (oy * oz);
    G[7]  = Jp[1];
    G[8]  = st * (-oy) + ct * (ox * oz);
    G[9]  = st * ( ox) + ct * (oy * oz);
    G[10] = 1.f + ct * (-(ox * ox + oy * oy));
    G[11] = Jp[2];
}

// ------ prep: K-pair interleaved M_eff:  Mi[(k>>1)*MPAD + m][0/1] -----------
// Mi element pair = { relu(A)*Kp at k=2*pair, same at k=2*pair+1 }.
// Coalesced reads (two consecutive Kp rows, contiguous m) and v2f writes.
__global__ void prep_meff(const float* __restrict__ A,
                          const float* __restrict__ Kp,
                          float* __restrict__ Mi) {
    size_t idx = (size_t)blockIdx.x * blockDim.x + threadIdx.x; // over (KPAD/2)*MPAD
    if (idx >= (size_t)(KPAD / 2) * MPAD) return;
    int pairidx = (int)(idx / MPAD);
    int m       = (int)(idx % MPAD);
    v2f out;
#pragma unroll
    for (int h = 0; h < 2; ++h) {
        int kidx = 2 * pairidx + h;
        float v = 0.f;
        if (kidx < KDIM && m < MDIM) {
            int jm = kidx / 5, f = kidx % 5;
            int n  = m / 3;
            float a = A[(size_t)jm * NV + n];
            a = a > 0.f ? a : 0.f;
            v = a * Kp[((size_t)jm * 5 + f) * MDIM + m];
        }
        out[h] = v;
    }
    *(v2f*)(Mi + idx * 2) = out;
}

// -------- WMMA f32 GEMM: Tp(200x20670) = Qpad(208x116) @ M_eff(116x20672) ---
// Each wave: one 16-row tile x 4 column tiles (64 cols), K-loop in steps of 4.
// A frag: one aligned b64 per lane (pitch 116*4=464 B, 8B aligned).
// B frag: one coalesced b64 per lane from interleaved Mi.
__global__ __launch_bounds__(32) void gemm_tp(const float* __restrict__ Qpad,
                                              const float* __restrict__ Mi,
                                              float* __restrict__ Tp) {
    const int tcg  = blockIdx.x;        // 0..MPAD/64-1 column group
    const int trow = blockIdx.y;        // 0..PPAD/16-1
    const int lane = threadIdx.x;
    const int half = lane >> 4;         // 0: K pair {0,1}, 1: K pair {2,3}
    const int l15  = lane & 15;
    const int arow = trow * 16 + l15;
    const int col0 = tcg * (16 * CT) + l15;
    v8f acc[CT];
#pragma unroll
    for (int t = 0; t < CT; ++t) acc[t] = (v8f){};
    for (int kk = 0; kk < KPAD; kk += 4) {
        const int ka = kk + half * 2;
        v2f a = *(const v2f*)(Qpad + (size_t)arow * KPAD + ka);
        const v2f* brow = (const v2f*)Mi + (size_t)(ka >> 1) * MPAD;
#pragma unroll
        for (int t = 0; t < CT; ++t) {
            v2f b = brow[col0 + t * 16];
            acc[t] = __builtin_amdgcn_wmma_f32_16x16x4_f32(
                /*neg_a=*/false, a, /*neg_b=*/false, b,
                /*c_mod=*/(short)0, acc[t], /*reuse_a=*/false, /*reuse_b=*/false);
        }
    }
    // D 16x16: VGPR r -> rows r (lanes 0-15) and r+8 (lanes 16-31)
#pragma unroll
    for (int t = 0; t < CT; ++t) {
        int outcol = col0 + t * 16;
        if (outcol < MDIM) {
#pragma unroll
            for (int r = 0; r < 8; ++r) {
                int orow = trow * 16 + r + half * 8;
                if (orow < PREG)
                    Tp[(size_t)orow * MDIM + outcol] = acc[t][r];
            }
        }
    }
}

// ---------------- skinning + E_D block partials ------------------------------
__global__ __launch_bounds__(256) void verts_k(const float* __restrict__ V,
                                               const float* __restrict__ T,
                                               const float* __restrict__ Tp,
                                               const float* __restrict__ Gmat,
                                               const float* __restrict__ W,
                                               float* __restrict__ outv,
                                               float* __restrict__ ed_part) {
    __shared__ float sG[KJ * 12];
    __shared__ float sred[256];
    const int p   = blockIdx.y;
    const int tid = threadIdx.x;
    const int n   = blockIdx.x * 256 + tid;
    for (int i = tid; i < KJ * 12; i += 256) sG[i] = Gmat[(size_t)p * KJ * 12 + i];
    __syncthreads();
    float dd = 0.f;
    if (n < NV) {
        size_t base = ((size_t)p * NV + n) * 3;
        float tc0 = T[base + 0] + Tp[(size_t)p * MDIM + 3 * n + 0];
        float tc1 = T[base + 1] + Tp[(size_t)p * MDIM + 3 * n + 1];
        float tc2 = T[base + 2] + Tp[(size_t)p * MDIM + 3 * n + 2];
        float M[12];
#pragma unroll
        for (int i = 0; i < 12; ++i) M[i] = 0.f;
        // W row is 24 contiguous floats, 16B aligned -> 6 x b128 loads
        const v4f* wr = (const v4f*)(W + (size_t)n * KJ);
#pragma unroll
        for (int q = 0; q < 6; ++q) {
            v4f w4 = wr[q];
#pragma unroll
            for (int c = 0; c < 4; ++c) {
                float w = w4[c];
                const float* g = sG + (4 * q + c) * 12;
#pragma unroll
                for (int i = 0; i < 12; ++i) M[i] = fmaf(w, g[i], M[i]);
            }
        }
        float v0 = fmaf(M[0], tc0, fmaf(M[1],  tc1, fmaf(M[2],  tc2, M[3])));
        float v1 = fmaf(M[4], tc0, fmaf(M[5],  tc1, fmaf(M[6],  tc2, M[7])));
        float v2 = fmaf(M[8], tc0, fmaf(M[9],  tc1, fmaf(M[10], tc2, M[11])));
        outv[base + 0] = v0;
        outv[base + 1] = v1;
        outv[base + 2] = v2;
        float d0 = V[base + 0] - v0;
        float d1 = V[base + 1] - v1;
        float d2 = V[base + 2] - v2;
        dd = d0 * d0 + d1 * d1 + d2 * d2;
    }
    sred[tid] = dd;
    __syncthreads();
    for (int s = 128; s > 0; s >>= 1) {
        if (tid < s) sred[tid] += sred[tid + s];
        __syncthreads();
    }
    if (tid == 0) ed_part[(size_t)p * gridDim.x + blockIdx.x] = sred[0];
}

// ---------------- per-joint ||K_params[j]||_F -------------------------------
__global__ __launch_bounds__(256) void ek_k(const float* __restrict__ Kp,
                                            float* __restrict__ ekj) {
    __shared__ float sred[256];
    const int j   = blockIdx.x;
    const int tid = threadIdx.x;
    const float* base = Kp + (size_t)j * 5 * MDIM;
    float s = 0.f;
    for (int i = tid; i < 5 * MDIM; i += 256) { float v = base[i]; s = fmaf(v, v, s); }
    sred[tid] = s;
    __syncthreads();
    for (int st = 128; st > 0; st >>= 1) {
        if (tid < st) sred[tid] += sred[tid + st];
        __syncthreads();
    }
    if (tid == 0) ekj[j] = sqrtf(sred[0]);
}

// ---------------- final deterministic combine -------------------------------
__global__ __launch_bounds__(256) void final_k(const float* __restrict__ W,
                                               const float* __restrict__ Wi,
                                               const float* __restrict__ A,
                                               const float* __restrict__ ed_part,
                                               int ned,
                                               const float* __restrict__ ekj,
                                               float* __restrict__ out) {
    __shared__ float sred[256];
    const int tid = threadIdx.x;
    float ed = 0.f;
    for (int i = tid; i < ned; i += 256) ed += ed_part[i];
    float ewi = 0.f, ew = 0.f;
    for (int i = tid; i < NV * KJ; i += 256) {
        float w = W[i];
        float d = w - Wi[i];
        ewi = fmaf(d, d, ewi);
        ew += fabsf(w);
    }
    float ea = 0.f;
    for (int i = tid; i < (KJ - 1) * NV; i += 256) ea += fabsf(A[i]);
    float ek = 0.f;
    for (int i = tid; i < KJ - 1; i += 256) ek += ekj[i];
    sred[tid] = 5.f * ed + ewi + 50.f * ew + 40.f * ea + 30.f * ek;
    __syncthreads();
    for (int s = 128; s > 0; s >>= 1) {
        if (tid < s) sred[tid] += sred[tid + s];
        __syncthreads();
    }
    if (tid == 0) out[0] = sred[0];
}

extern "C" void kernel_launch(void* const* d_in, const int* in_sizes, int n_in,
                              void* d_out, int out_size, void* d_ws, size_t ws_size,
                              hipStream_t stream) {
    const float* V     = (const float*)d_in[0];
    const float* T     = (const float*)d_in[1];
    const float* J     = (const float*)d_in[2];
    const float* theta = (const float*)d_in[3];
    const float* beta2 = (const float*)d_in[4];
    const float* W     = (const float*)d_in[5];
    const float* Wi    = (const float*)d_in[6];
    const float* A     = (const float*)d_in[7];
    const float* Kp    = (const float*)d_in[8];
    float* out = (float*)d_out;

    // workspace layout (floats): ~26.5 MB total
    float* ws   = (float*)d_ws;
    float* Qpad = ws;                                   // PPAD*KPAD   = 24128
    float* Mi   = Qpad + (size_t)PPAD * KPAD;           // KPAD*MPAD   = 2397952
    float* Tp   = Mi   + (size_t)KPAD * MPAD;           // PREG*MDIM   = 4134000
    float* Gmat = Tp   + (size_t)PREG * MDIM;           // PREG*KJ*12  = 57600
    float* edp  = Gmat + (size_t)PREG * KJ * 12;        // 200*27      = 5400
    float* ekj  = edp  + (size_t)PREG * 27;             // 23

    prep_q  <<<(PPAD * KJ + 255) / 256, 256, 0, stream>>>(theta, beta2, Qpad);
    prep_rod<<<(PREG * KJ + 255) / 256, 256, 0, stream>>>(theta, J, Gmat);
    prep_meff<<<(int)(((size_t)(KPAD / 2) * MPAD + 255) / 256), 256, 0, stream>>>(A, Kp, Mi);

    dim3 gg(MPAD / (16 * CT), PPAD / 16);               // 323 x 13 tile groups
    gemm_tp<<<gg, 32, 0, stream>>>(Qpad, Mi, Tp);

    dim3 gv((NV + 255) / 256, PREG);                    // 27 x 200
    verts_k<<<gv, 256, 0, stream>>>(V, T, Tp, Gmat, W, out + 1, edp);

    ek_k   <<<KJ - 1, 256, 0, stream>>>(Kp, ekj);
    final_k<<<1, 256, 0, stream>>>(W, Wi, A, edp, PREG * 27, ekj, out);
}